// PhysicsConstraintLoss_4063039062201
// MI455X (gfx1250) — compile-verified
//
#include <hip/hip_runtime.h>
#include <math.h>

// Problem constants (from reference setup_inputs)
#define NROWS 131072
#define LLEN  256     // K dimension
#define DDIM  128
#define CCLS  64
#define NBAND 4
#define NCOL  384     // D + 128 cos + 128 sin
#define EPSF  1e-8f
#define ASTRIDE 258   // padded LDS stride (words) to kill bank conflicts

typedef float v2f __attribute__((ext_vector_type(2)));
typedef float v4f __attribute__((ext_vector_type(4)));
typedef float v8f __attribute__((ext_vector_type(8)));

// ---------------------------------------------------------------- zero scratch
__global__ void zero_ws_kernel(float* __restrict__ p, int n) {
    int i = blockIdx.x * blockDim.x + threadIdx.x;
    if (i < n) p[i] = 0.0f;
}

// -------------------------------------- build transposed B^T = [Wphi|cos|sin]^T
// Bt is [NCOL][LLEN] row-major (i.e., column-major view of the K x N operand),
// so a WMMA B-fragment's (kk, kk+1) pair is contiguous -> single b64 load.
// Rows 0..127 = W_phi column d, 128..255 = cos(2πkt/L), 256..383 = sin(2πkt/L).
__global__ void build_Bt_kernel(const float* __restrict__ Wphi, float* __restrict__ Bt) {
    int idx = blockIdx.x * blockDim.x + threadIdx.x;   // 0 .. NCOL*LLEN-1
    int d = idx / LLEN;       // output column (N index)
    int t = idx - d * LLEN;   // K index
    float val;
    if (d < DDIM) {
        val = Wphi[t * DDIM + d];
    } else {
        int k  = (d < 2 * DDIM) ? (d - DDIM) : (d - 2 * DDIM);
        int ph = (k * t) & (LLEN - 1);                 // exact phase mod L
        float ang = (float)ph * (6.28318530717958647692f / (float)LLEN);
        val = (d < 2 * DDIM) ? __cosf(ang) : __sinf(ang);
    }
    Bt[idx] = val;
}

// ---------------------------------------------------------------- fused main
// grid.x = NROWS/16 row tiles, block = 256 threads (8 waves).
// Each wave computes 3 WMMA 16x16 output tiles: feats, cos-dot, sin-dot.
__global__ __launch_bounds__(256) void fused_main_kernel(
    const float* __restrict__ x, const int* __restrict__ y,
    const float* __restrict__ Bt, const float* __restrict__ v,
    const float* __restrict__ Etgt,
    float* __restrict__ sums, float* __restrict__ counts,
    float* __restrict__ lossE)
{
    __shared__ float Atile[16 * ASTRIDE];
    __shared__ float bandAcc[16 * NBAND];
    __shared__ int   ycls[16];

    const int tid     = threadIdx.x;
    const int lane    = tid & 31;
    const int wave    = tid >> 5;
    const int rowbase = blockIdx.x * 16;

    if (tid < 16) ycls[tid] = y[rowbase + tid];
    if (tid < 16 * NBAND) bandAcc[tid] = 0.0f;

    // Cooperative A-tile load: 16 rows x 256 = 4096 floats, 16 per thread.
    // x_gen is streamed once over the whole grid -> non-temporal loads so the
    // 134 MB stream doesn't evict the hot 384 KB B matrix from L2.
    {
        int f = tid * 16;
        int r = f >> 8;
        int c = f & 255;
        const v4f* src = (const v4f*)(x + (size_t)(rowbase + r) * LLEN + c);
        float* dst = &Atile[r * ASTRIDE + c];
        #pragma unroll
        for (int i = 0; i < 4; ++i) {
            v4f q = __builtin_nontemporal_load(src + i);
            dst[i * 4 + 0] = q.x; dst[i * 4 + 1] = q.y;
            dst[i * 4 + 2] = q.z; dst[i * 4 + 3] = q.w;
        }
    }
    __syncthreads();

    v8f accF = {}; v8f accC = {}; v8f accS = {};

    const int n    = lane & 15;          // N-index within tile / A row
    const int khi  = (lane >> 4) << 1;   // K split: lanes 16-31 hold K+2,K+3
    // Per-lane base pointers into transposed B for the wave's three tiles.
    const float* btF = Bt + (size_t)(wave * 16 + n) * LLEN;            // feats
    const float* btC = Bt + (size_t)(DDIM + wave * 16 + n) * LLEN;     // cos
    const float* btS = Bt + (size_t)(2 * DDIM + wave * 16 + n) * LLEN; // sin

    #pragma unroll 4
    for (int k0 = 0; k0 < LLEN; k0 += 4) {
        const int kk = k0 + khi;
        // A fragment: lane<16 -> (K=k0,k0+1) of row n; lane>=16 -> (k0+2,k0+3)
        const float* ap = &Atile[n * ASTRIDE + kk];
        v2f a; a.x = ap[0]; a.y = ap[1];
        // B fragments: contiguous (kk, kk+1) pair -> single 64-bit load each
        v2f bF = *(const v2f*)(btF + kk);
        v2f bC = *(const v2f*)(btC + kk);
        v2f bS = *(const v2f*)(btS + kk);

        accF = __builtin_amdgcn_wmma_f32_16x16x4_f32(false, a, false, bF, (short)0, accF, false, false);
        accC = __builtin_amdgcn_wmma_f32_16x16x4_f32(false, a, false, bC, (short)0, accC, false, false);
        accS = __builtin_amdgcn_wmma_f32_16x16x4_f32(false, a, false, bS, (short)0, accS, false, false);
    }

    // C/D layout: VGPR j, lanes 0-15 -> M=j, N=lane; lanes 16-31 -> M=j+8.
    const int mbase = (lane >> 4) << 3;

    // ---- feats tile -> per-class variance partial sums (global f32 atomics)
    #pragma unroll
    for (int j = 0; j < 8; ++j) {
        int row16 = j + mbase;
        int cls   = ycls[row16];
        int dcol  = wave * 16 + n;
        float diff = accF[j] - v[cls * DDIM + dcol];
        atomicAdd(&sums[cls * DDIM + dcol], diff * diff);
    }

    // ---- band energy: this wave's 16 k-values lie entirely in band wave>>1
    #pragma unroll
    for (int j = 0; j < 8; ++j) {
        float p = accC[j] * accC[j] + accS[j] * accS[j];
        #pragma unroll
        for (int off = 1; off < 16; off <<= 1)
            p += __shfl_xor(p, off, 32);
        if (n == 0)
            atomicAdd(&bandAcc[(j + mbase) * NBAND + (wave >> 1)], p);
    }
    __syncthreads();

    // ---- per-row ratio + L1 against E_target[y], and class counts
    if (tid < 16) {
        float b0 = bandAcc[tid * NBAND + 0];
        float b1 = bandAcc[tid * NBAND + 1];
        float b2 = bandAcc[tid * NBAND + 2];
        float b3 = bandAcc[tid * NBAND + 3];
        float inv = 1.0f / (b0 + b1 + b2 + b3 + EPSF);
        int cls = ycls[tid];
        const float* et = Etgt + cls * NBAND;
        float le = fabsf(b0 * inv - et[0]) + fabsf(b1 * inv - et[1]) +
                   fabsf(b2 * inv - et[2]) + fabsf(b3 * inv - et[3]);
        atomicAdd(lossE, le);
        atomicAdd(&counts[cls], 1.0f);
    }
}

// ---------------------------------------------------------------- finalize
__global__ __launch_bounds__(64) void finalize_kernel(
    const float* __restrict__ sums, const float* __restrict__ counts,
    const float* __restrict__ lossE, const float* __restrict__ emaVar,
    const float* __restrict__ wTgt, float* __restrict__ out)
{
    __shared__ float redW[64];
    __shared__ float redP[64];
    const int c = threadIdx.x;   // one class per thread

    float cnt = counts[c];
    bool present = cnt > 0.0f;
    float dn = fmaxf(cnt, 1.0f);

    float s = 0.0f;
    for (int d = 0; d < DDIM; ++d) {
        float var = present ? (sums[c * DDIM + d] / dn) : emaVar[c * DDIM + d];
        s += rsqrtf(var + EPSF);
    }
    float sinv = 1.0f / (s + EPSF);
    float mse = 0.0f;
    for (int d = 0; d < DDIM; ++d) {
        float var = present ? (sums[c * DDIM + d] / dn) : emaVar[c * DDIM + d];
        float wg = rsqrtf(var + EPSF) * sinv;
        float df = wg - wTgt[c * DDIM + d];
        mse += df * df;
    }
    mse *= (1.0f / (float)DDIM);

    redW[c] = present ? mse : 0.0f;
    redP[c] = present ? 1.0f : 0.0f;
    __syncthreads();
    for (int off = 32; off > 0; off >>= 1) {
        if (c < off) { redW[c] += redW[c + off]; redP[c] += redP[c + off]; }
        __syncthreads();
    }
    if (c == 0) {
        float lw = redW[0] / fmaxf(redP[0], 1.0f);
        float le = lossE[0] / (float)(NROWS * NBAND);
        out[0] = lw + le;   // total
        out[1] = lw;        // loss_w
        out[2] = le;        // loss_E
    }
}

// ---------------------------------------------------------------- launcher
extern "C" void kernel_launch(void* const* d_in, const int* in_sizes, int n_in,
                              void* d_out, int out_size, void* d_ws, size_t ws_size,
                              hipStream_t stream) {
    const float* x      = (const float*)d_in[0];   // [N, L]
    const int*   y      = (const int*)  d_in[1];   // [N]
    const float* Wphi   = (const float*)d_in[2];   // [L, D]
    const float* v      = (const float*)d_in[3];   // [C, D]
    const float* wTgt   = (const float*)d_in[4];   // [C, D]
    const float* eTgt   = (const float*)d_in[5];   // [C, 4]
    const float* emaVar = (const float*)d_in[6];   // [C, D]

    float* ws     = (float*)d_ws;
    float* Bmat   = ws;                         // NCOL*LLEN = 98304 floats (B^T)
    float* sums   = Bmat + NCOL * LLEN;         // C*D = 8192 floats
    float* counts = sums + CCLS * DDIM;         // 64 floats
    float* lossE  = counts + CCLS;              // 1 float

    const int nzero = CCLS * DDIM + CCLS + 1;
    zero_ws_kernel<<<(nzero + 255) / 256, 256, 0, stream>>>(sums, nzero);
    build_Bt_kernel<<<(NCOL * LLEN) / 256, 256, 0, stream>>>(Wphi, Bmat);
    fused_main_kernel<<<NROWS / 16, 256, 0, stream>>>(x, y, Bmat, v, eTgt,
                                                      sums, counts, lossE);
    finalize_kernel<<<1, 64, 0, stream>>>(sums, counts, lossE, emaVar, wTgt,
                                          (float*)d_out);
}